// RetentiveAttention_24927990186154
// MI455X (gfx1250) — compile-verified
//
#include <hip/hip_runtime.h>
#include <hip/hip_bf16.h>

#define L_CFG 4
#define G_CFG 20000
#define C_CFG 128
#define E_CFG 320000
#define KD_CFG 16
#define ROWS (G_CFG * L_CFG) /* 80000 */
#define DECAY_F 0.5f
#define LN_EPS_F 1e-5f

#define CHUNK 256
#define NCHUNK ((G_CFG + CHUNK - 1) / CHUNK) /* 79 */

typedef __attribute__((ext_vector_type(2))) float v2f;
typedef __attribute__((ext_vector_type(8))) float v8f;

// ---------------------------------------------------------------------------
// K1: x[L,G,C] -> row-major [G*L, 128] (row r = g*4 + l); xacc = same; wsum=0
// ---------------------------------------------------------------------------
__global__ void k_init(const float* __restrict__ x, float* __restrict__ cur,
                       float* __restrict__ xacc, float* __restrict__ wsum) {
  long idx = (long)blockIdx.x * blockDim.x + threadIdx.x;
  const long total = (long)L_CFG * G_CFG * C_CFG;
  if (idx < total) {
    int l = (int)(idx / ((long)G_CFG * C_CFG));
    int rem = (int)(idx % ((long)G_CFG * C_CFG));
    int g = rem / C_CFG;
    int c = rem % C_CFG;
    float val = x[idx];
    long dst = (long)g * (L_CFG * C_CFG) + l * C_CFG + c;
    cur[dst] = val;
    xacc[dst] = val;
  }
  if (idx < ROWS) wsum[idx] = 0.0f;
}

// ---------------------------------------------------------------------------
// CSR construction: histogram -> chunked exclusive scan -> slot fill.
// Replaces 328M scatter atomics with ~0.7M small int atomics.
// ---------------------------------------------------------------------------
__global__ void k_hist_zero(int* __restrict__ counts) {
  int i = blockIdx.x * blockDim.x + threadIdx.x;
  if (i < G_CFG) counts[i] = 0;
}

__global__ void k_hist(const int* __restrict__ erow, int* __restrict__ counts) {
  int e = blockIdx.x * blockDim.x + threadIdx.x;
  if (e < E_CFG) atomicAdd(&counts[erow[e]], 1);
}

__global__ void k_scan1(const int* __restrict__ counts, int* __restrict__ localExcl,
                        int* __restrict__ chunkSum) {
  __shared__ int s[CHUNK];
  int b = blockIdx.x, t = threadIdx.x;
  int i = b * CHUNK + t;
  int v = (i < G_CFG) ? counts[i] : 0;
  s[t] = v;
  __syncthreads();
  for (int off = 1; off < CHUNK; off <<= 1) {   // Hillis-Steele inclusive scan
    int add = (t >= off) ? s[t - off] : 0;
    __syncthreads();
    s[t] += add;
    __syncthreads();
  }
  if (i < G_CFG) localExcl[i] = s[t] - v;
  if (t == CHUNK - 1) chunkSum[b] = s[t];
}

__global__ void k_scan2(const int* __restrict__ chunkSum, int* __restrict__ chunkOff) {
  if (threadIdx.x == 0) {
    int acc = 0;
    for (int b = 0; b < NCHUNK; ++b) { chunkOff[b] = acc; acc += chunkSum[b]; }
  }
}

__global__ void k_scan3(const int* __restrict__ localExcl, const int* __restrict__ chunkOff,
                        int* __restrict__ rowptr, int* __restrict__ cursor) {
  int i = blockIdx.x * blockDim.x + threadIdx.x;
  if (i < G_CFG) {
    int v = localExcl[i] + chunkOff[i / CHUNK];
    rowptr[i] = v;
    cursor[i] = v;
  }
  if (i == 0) rowptr[G_CFG] = E_CFG;
}

__global__ void k_fill(const int* __restrict__ erow, int* __restrict__ cursor,
                       int* __restrict__ eidx) {
  int e = blockIdx.x * blockDim.x + threadIdx.x;
  if (e < E_CFG) {
    int slot = atomicAdd(&cursor[erow[e]], 1);
    eidx[slot] = e;
  }
}

// ---------------------------------------------------------------------------
// Gather SpMM: dst[g,:] = sum_e (DECAY*val_e) * src[col_e,:]  -- no atomics.
// One block per row g, 128 threads x float4 (512 features). Edge metadata
// staged through LDS; next neighbor row prefetched (global_prefetch_b8).
// ---------------------------------------------------------------------------
__global__ void k_spmm_gather(const int* __restrict__ rowptr, const int* __restrict__ eidx,
                              const int* __restrict__ ecol, const float* __restrict__ evals,
                              const float* __restrict__ src, float* __restrict__ dst) {
  __shared__ int   s_col[128];
  __shared__ float s_val[128];
  int g = blockIdx.x;
  int t = threadIdx.x; // 0..127
  int beg = rowptr[g];
  int end = rowptr[g + 1];
  float4 acc = {0.0f, 0.0f, 0.0f, 0.0f};
  for (int base = beg; base < end; base += 128) {
    int n = end - base;
    if (n > 128) n = 128;
    if (t < n) {
      int e = eidx[base + t];
      s_col[t] = ecol[e];
      s_val[t] = evals[e] * DECAY_F;
    }
    __syncthreads();
    for (int j = 0; j < n; ++j) {
      if (j + 1 < n)
        __builtin_prefetch(src + (long)s_col[j + 1] * 512 + t * 4, 0, 3);
      const float4* s4 = (const float4*)(src + (long)s_col[j] * 512);
      float4 d = s4[t];
      float v = s_val[j];
      acc.x += v * d.x;
      acc.y += v * d.y;
      acc.z += v * d.z;
      acc.w += v * d.w;
    }
    __syncthreads();
  }
  float4* o = (float4*)(dst + (long)g * 512);
  o[t] = acc;   // full row written, including zero-degree rows -> no pre-zero
}

// ---------------------------------------------------------------------------
// xacc += src
// ---------------------------------------------------------------------------
__global__ void k_accadd(float* __restrict__ xacc, const float* __restrict__ src, long n) {
  long idx = (long)blockIdx.x * blockDim.x + threadIdx.x;
  if (idx < n) xacc[idx] += src[idx];
}

// ---------------------------------------------------------------------------
// wsum[r] += mean_d( (X@Wk)[r,d] * (X@Wq)[r,d] )  via WMMA f32 16x16x4.
// One wave per 16-row tile; shared A fragment for the k- and q-GEMM.
// ---------------------------------------------------------------------------
__global__ void k_kq(const float* __restrict__ X, const float* __restrict__ Wk,
                     const float* __restrict__ Wq, float* __restrict__ wsum) {
  __shared__ float prod[16][17];
  int t = threadIdx.x;        // 0..31, all lanes active (EXEC all-ones)
  int r0 = blockIdx.x * 16;
  int nl = t & 15;
  int kh = t >> 4;
  const float* arow = X + (long)(r0 + nl) * C_CFG;

  v8f kacc = {};
  v8f qacc = {};
  for (int kk = 0; kk < 32; ++kk) {
    int kb = kk * 4 + kh * 2;
    v2f a;  a.x = arow[kb];              a.y = arow[kb + 1];
    v2f bk; bk.x = Wk[kb * KD_CFG + nl]; bk.y = Wk[(kb + 1) * KD_CFG + nl];
    v2f bq; bq.x = Wq[kb * KD_CFG + nl]; bq.y = Wq[(kb + 1) * KD_CFG + nl];
    kacc = __builtin_amdgcn_wmma_f32_16x16x4_f32(false, a, false, bk,
                                                 (short)0, kacc, false, false);
    qacc = __builtin_amdgcn_wmma_f32_16x16x4_f32(false, a, false, bq,
                                                 (short)0, qacc, false, false);
  }
#pragma unroll
  for (int v = 0; v < 8; ++v)
    prod[v + 8 * kh][nl] = kacc[v] * qacc[v];
  __syncthreads();
  if (t < 16) {
    float s = 0.0f;
#pragma unroll
    for (int n = 0; n < KD_CFG; ++n) s += prod[t][n];
    wsum[r0 + t] += s * (1.0f / KD_CFG);
  }
}

// ---------------------------------------------------------------------------
// values = Xacc @ Wv, then v = values*w, LayerNorm over C, write [L,G,C].
// 256 threads = 8 wave32s per block; Wv (64KB) staged once into LDS with
// coalesced b128 loads; after accumulation the LDS is reused (stride 132,
// bank-conflict-free) as the 128x128 output staging tile for the epilogue.
// ---------------------------------------------------------------------------
#define TSTRIDE 132
__global__ void __launch_bounds__(256) k_gemm_ln(
    const float* __restrict__ Xacc, const float* __restrict__ Wv,
    const float* __restrict__ wsum, const float* __restrict__ gamma,
    const float* __restrict__ beta, float* __restrict__ out) {
  __shared__ __align__(16) float smem[8 * 16 * TSTRIDE]; // 16896 floats = 66KB
  int t = threadIdx.x;        // 0..255
  int wave = t >> 5;          // 0..7
  int lane = t & 31;

  // stage Wv: 16384 floats, 4096 float4s, 16 iters x 256 threads
  {
    float4* s4 = (float4*)smem;
    const float4* w4 = (const float4*)Wv;
    for (int i = t; i < (C_CFG * C_CFG) / 4; i += 256) s4[i] = w4[i];
  }
  __syncthreads();

  int r0 = blockIdx.x * 128 + wave * 16;
  int nl = lane & 15;
  int kh = lane >> 4;
  const float* arow = Xacc + (long)(r0 + nl) * C_CFG;

  v8f acc[8];
#pragma unroll
  for (int j = 0; j < 8; ++j) { v8f z = {}; acc[j] = z; }

  for (int kk = 0; kk < 32; ++kk) {
    int kb = kk * 4 + kh * 2;
    v2f a; a.x = arow[kb]; a.y = arow[kb + 1];
#pragma unroll
    for (int j = 0; j < 8; ++j) {
      int n = j * 16 + nl;
      v2f b; b.x = smem[kb * C_CFG + n]; b.y = smem[(kb + 1) * C_CFG + n];
      acc[j] = __builtin_amdgcn_wmma_f32_16x16x4_f32(false, a, false, b,
                                                     (short)0, acc[j], false, false);
    }
  }
  __syncthreads();   // everyone done reading Wv from LDS

  // repurpose LDS as per-wave 16 x 128 tiles (stride 132)
  float* tile = smem + wave * 16 * TSTRIDE;
#pragma unroll
  for (int j = 0; j < 8; ++j)
#pragma unroll
    for (int v = 0; v < 8; ++v)
      tile[(v + 8 * kh) * TSTRIDE + j * 16 + nl] = acc[j][v];
  __syncthreads();

  if (lane < 16) {
    int r = r0 + lane;
    float w = wsum[r];
    const float* trow = tile + lane * TSTRIDE;
    float mu = 0.0f;
    for (int c = 0; c < C_CFG; ++c) mu += trow[c] * w;
    mu *= (1.0f / C_CFG);
    float var = 0.0f;
    for (int c = 0; c < C_CFG; ++c) {
      float d = trow[c] * w - mu;
      var += d * d;
    }
    var *= (1.0f / C_CFG);
    float inv = rsqrtf(var + LN_EPS_F);
    int l = r & 3;
    int g = r >> 2;
    float* o = out + (long)l * G_CFG * C_CFG + (long)g * C_CFG;
    for (int c = 0; c < C_CFG; ++c)
      o[c] = (trow[c] * w - mu) * inv * gamma[c] + beta[c];
  }
}

// ---------------------------------------------------------------------------
extern "C" void kernel_launch(void* const* d_in, const int* in_sizes, int n_in,
                              void* d_out, int out_size, void* d_ws, size_t ws_size,
                              hipStream_t stream) {
  const float* x     = (const float*)d_in[0];
  const int*   erow  = (const int*)d_in[1];
  const int*   ecol  = (const int*)d_in[2];
  const float* evals = (const float*)d_in[3];
  const float* Wk    = (const float*)d_in[4];
  const float* Wq    = (const float*)d_in[5];
  const float* Wv    = (const float*)d_in[6];
  const float* gamma = (const float*)d_in[7];
  const float* beta  = (const float*)d_in[8];
  float* out = (float*)d_out;

  const long NTOT = (long)L_CFG * G_CFG * C_CFG; // 10,240,000
  float* bufA = (float*)d_ws;
  float* bufB = bufA + NTOT;
  float* xacc = bufB + NTOT;
  float* wsum = xacc + NTOT;
  int* counts    = (int*)(wsum + ROWS);
  int* localExcl = counts + G_CFG;
  int* chunkSum  = localExcl + G_CFG;
  int* chunkOff  = chunkSum + NCHUNK;
  int* rowptr    = chunkOff + NCHUNK;
  int* cursor    = rowptr + (G_CFG + 1);
  int* eidx      = cursor + G_CFG;

  const int B = 256;
  const unsigned gN = (unsigned)((NTOT + B - 1) / B);
  const unsigned gG = (G_CFG + B - 1) / B;
  const unsigned gE = (E_CFG + B - 1) / B;
  const unsigned gTiles = ROWS / 16; // 5000

  // init + CSR build
  k_init<<<gN, B, 0, stream>>>(x, bufA, xacc, wsum);
  k_hist_zero<<<gG, B, 0, stream>>>(counts);
  k_hist<<<gE, B, 0, stream>>>(erow, counts);
  k_scan1<<<NCHUNK, CHUNK, 0, stream>>>(counts, localExcl, chunkSum);
  k_scan2<<<1, 32, 0, stream>>>(chunkSum, chunkOff);
  k_scan3<<<gG, B, 0, stream>>>(localExcl, chunkOff, rowptr, cursor);
  k_fill<<<gE, B, 0, stream>>>(erow, cursor, eidx);

  // iteration-0 weights
  k_kq<<<gTiles, 32, 0, stream>>>(bufA, Wk, Wq, wsum);

  // iteration 1: bufB = spmm(DECAY * bufA)
  k_spmm_gather<<<G_CFG, 128, 0, stream>>>(rowptr, eidx, ecol, evals, bufA, bufB);
  k_kq<<<gTiles, 32, 0, stream>>>(bufB, Wk, Wq, wsum);
  k_accadd<<<gN, B, 0, stream>>>(xacc, bufB, NTOT);

  // iteration 2: bufA = spmm(DECAY * bufB)
  k_spmm_gather<<<G_CFG, 128, 0, stream>>>(rowptr, eidx, ecol, evals, bufB, bufA);
  k_kq<<<gTiles, 32, 0, stream>>>(bufA, Wk, Wq, wsum);
  k_accadd<<<gN, B, 0, stream>>>(xacc, bufA, NTOT);

  // values = xacc @ Wv ; scale ; LayerNorm ; write [L,G,C]
  k_gemm_ln<<<ROWS / 128, 256, 0, stream>>>(xacc, Wv, wsum, gamma, beta, out);
}